// NodeViewReadoutFFN_9964324127439
// MI455X (gfx1250) — compile-verified
//
#include <hip/hip_runtime.h>
#include <hip/hip_bf16.h>

// ---------------- problem constants (match reference) ----------------
#define N_ATOMS   100000
#define N_MOLS    2500
#define MAX_NB    10
#define HID       256
#define ATOM_FDIM 151
#define FEAT_DIM  200
#define FFN_HID   1024
#define N_TASKS   12
#define LN_EPS    1e-5f

#define D_IN      (HID + ATOM_FDIM)   // 407
#define D_FF      (4 * HID)           // 1024
#define D_MOL_IN  (HID + FEAT_DIM)    // 456

#define K1P       416                 // D_IN padded to multiple of 16
#define KMOLP     464                 // D_MOL_IN padded to multiple of 16

typedef float v2f __attribute__((ext_vector_type(2)));
typedef float v8f __attribute__((ext_vector_type(8)));

__device__ __forceinline__ float4 ld4(const float* p) {
    return *reinterpret_cast<const float4*>(p);
}
__device__ __forceinline__ void st4(float* p, float4 v) {
    *reinterpret_cast<float4*>(p) = v;
}

// ---------------- pad weight rows with zeros (K -> Kp) ----------------
__global__ void pad_rows_kernel(const float* __restrict__ src,
                                float* __restrict__ dst,
                                int K, int Kp, int N) {
    size_t idx = (size_t)blockIdx.x * blockDim.x + threadIdx.x;
    size_t total = (size_t)Kp * N;
    if (idx >= total) return;
    int r = (int)(idx / N);
    int c = (int)(idx % N);
    dst[idx] = (r < K) ? src[(size_t)r * N + c] : 0.0f;
}

// ---------------- fused neighbor-gather-sum + concat + pad ----------------
// x[atom] = [ f_atoms[atom] (151) | sum_j atom_output[a2a[atom][j]] (256) | 0 (9) ]
__global__ __launch_bounds__(HID) void gather_concat_kernel(
        const float* __restrict__ atom_output,
        const float* __restrict__ f_atoms,
        const int*   __restrict__ a2a,
        float* __restrict__ xbuf) {
    int atom = blockIdx.x;
    int tid  = threadIdx.x;                    // 256 threads

    int nb[MAX_NB];
    #pragma unroll
    for (int j = 0; j < MAX_NB; ++j)           // uniform per block -> SGPR loads
        nb[j] = a2a[(size_t)atom * MAX_NB + j];

    float s = 0.0f;
    #pragma unroll
    for (int j = 0; j < MAX_NB; ++j)
        s += atom_output[(size_t)nb[j] * HID + tid];

    float* xrow = xbuf + (size_t)atom * K1P;
    xrow[ATOM_FDIM + tid] = s;                                 // cols 151..406
    if (tid < ATOM_FDIM)
        xrow[tid] = f_atoms[(size_t)atom * ATOM_FDIM + tid];   // cols 0..150
    if (tid < K1P - D_IN)
        xrow[D_IN + tid] = 0.0f;                               // cols 407..415
}

// ---------------- generic tiled f32 WMMA GEMM: C = act(A @ B + bias) ----------------
// A: [M,K] (K multiple of 16), B: [K,N] (N multiple of 256), C: [M,N]
// Block: 256 threads = 8 waves; block tile 32x256; wave tile 16x64 (4 v8f accs).
// Double-buffered LDS staging with b128 global loads / LDS stores.
// v_wmma_f32_16x16x4_f32 fragment layout (ISA 7.12.2):
//   A 16x4 : lane m=lane&15, half h=lane>>4: {A[m][2h], A[m][2h+1]}
//   B 4x16 : lane n=lane&15:                  {B[2h][n], B[2h+1][n]}
//   C 16x16: vgpr v: row = v + 8h, col = lane&15
template <bool RELU>
__global__ __launch_bounds__(256) void gemm_wmma_kernel(
        const float* __restrict__ A, const float* __restrict__ B,
        const float* __restrict__ bias, float* __restrict__ C,
        int M, int N, int K) {
    // stride 20 floats: 16B-aligned rows; frag reads bank-conflict-free
    __shared__ float As[2][32][20];
    // stride 272 floats: 16B-aligned rows; half-waves on disjoint bank groups
    __shared__ float Bs[2][16][272];

    int tid  = threadIdx.x;
    int lane = tid & 31;
    int w    = tid >> 5;            // 0..7
    int wm   = w >> 2;              // 0..1  (M sub-tile)
    int wn   = w & 3;               // 0..3  (N sub-tile, 64 cols each)
    int hh   = lane >> 4;           // half-wave
    int l15  = lane & 15;

    int blockM = blockIdx.x * 32;
    int blockN = blockIdx.y * 256;

    // A staging: threads 0..127 each load one float4 of the 32x16 tile
    int ar  = tid >> 2;             // 0..31 (row in tile)
    int ac  = (tid & 3) << 2;       // 0,4,8,12 (col)
    int arow = blockM + ar;

    v8f acc[4] = {};
    float4 ra = make_float4(0.f, 0.f, 0.f, 0.f);
    float4 rb[4];

    int NC = K >> 4;

    // ---- prologue: fetch chunk 0 into registers, store to LDS buf 0 ----
    if (tid < 128)
        ra = (arow < M) ? ld4(A + (size_t)arow * K + ac)
                        : make_float4(0.f, 0.f, 0.f, 0.f);
    #pragma unroll
    for (int i = 0; i < 4; ++i) {
        int idx = tid + i * 256;            // 1024 float4 of 16x256 tile
        int r = idx >> 6, c = (idx & 63) << 2;
        rb[i] = ld4(B + (size_t)r * N + blockN + c);
    }
    if (tid < 128) st4(&As[0][ar][ac], ra);
    #pragma unroll
    for (int i = 0; i < 4; ++i) {
        int idx = tid + i * 256;
        int r = idx >> 6, c = (idx & 63) << 2;
        st4(&Bs[0][r][c], rb[i]);
    }

    for (int ch = 0; ch < NC; ++ch) {
        __syncthreads();                    // staging of chunk `ch` complete
        int buf  = ch & 1;
        bool more = (ch + 1) < NC;
        int k0n  = (ch + 1) << 4;

        // ---- issue next chunk's global loads (overlap with WMMAs) ----
        if (more) {
            if (tid < 128)
                ra = (arow < M) ? ld4(A + (size_t)arow * K + k0n + ac)
                                : make_float4(0.f, 0.f, 0.f, 0.f);
            #pragma unroll
            for (int i = 0; i < 4; ++i) {
                int idx = tid + i * 256;
                int r = idx >> 6, c = (idx & 63) << 2;
                rb[i] = ld4(B + (size_t)(k0n + r) * N + blockN + c);
            }
        }

        // ---- compute chunk `ch` from LDS ----
        #pragma unroll
        for (int kk = 0; kk < 4; ++kk) {
            int kb = kk * 4 + 2 * hh;
            v2f a;
            a.x = As[buf][wm * 16 + l15][kb];
            a.y = As[buf][wm * 16 + l15][kb + 1];
            #pragma unroll
            for (int nt = 0; nt < 4; ++nt) {
                int nb_ = wn * 64 + nt * 16 + l15;
                v2f b;
                b.x = Bs[buf][kb][nb_];
                b.y = Bs[buf][kb + 1][nb_];
                acc[nt] = __builtin_amdgcn_wmma_f32_16x16x4_f32(
                    false, a, false, b, (short)0, acc[nt], false, false);
            }
        }

        // ---- store next chunk into the other LDS buffer ----
        if (more) {
            int nbuf = buf ^ 1;
            if (tid < 128) st4(&As[nbuf][ar][ac], ra);
            #pragma unroll
            for (int i = 0; i < 4; ++i) {
                int idx = tid + i * 256;
                int r = idx >> 6, c = (idx & 63) << 2;
                st4(&Bs[nbuf][r][c], rb[i]);
            }
        }
    }

    // ---- epilogue: bias (+ReLU) and store ----
    #pragma unroll
    for (int nt = 0; nt < 4; ++nt) {
        int col = blockN + wn * 64 + nt * 16 + l15;
        float bv = bias[col];
        #pragma unroll
        for (int v = 0; v < 8; ++v) {
            int row = blockM + wm * 16 + v + 8 * hh;
            if (row < M) {
                float val = acc[nt][v] + bv;
                if (RELU) val = fmaxf(val, 0.0f);
                C[(size_t)row * N + col] = val;
            }
        }
    }
}

// ---------------- per-row LayerNorm + per-molecule mean + concat features ----------------
// One block (256 threads) per molecule; thread tid owns column tid of HID.
__global__ __launch_bounds__(HID) void ln_segmean_kernel(
        const float* __restrict__ hpre, const int* __restrict__ a_scope,
        const float* __restrict__ ln_g, const float* __restrict__ ln_b,
        const float* __restrict__ features, float* __restrict__ molbuf) {
    int m   = blockIdx.x;
    int tid = threadIdx.x;
    int start = a_scope[m * 2 + 0];
    int size  = a_scope[m * 2 + 1];

    __shared__ float red[8];
    float g  = ln_g[tid];
    float bb = ln_b[tid];
    float colsum = 0.0f;

    for (int j = 0; j < size; ++j) {
        float v = hpre[(size_t)(start + j) * HID + tid];

        float s = v;                                    // mean reduction
        #pragma unroll
        for (int off = 16; off > 0; off >>= 1) s += __shfl_xor(s, off, 32);
        if ((tid & 31) == 0) red[tid >> 5] = s;
        __syncthreads();
        float tot = red[0] + red[1] + red[2] + red[3]
                  + red[4] + red[5] + red[6] + red[7];
        float mu = tot * (1.0f / HID);

        float d  = v - mu;                              // variance reduction
        float s2 = d * d;
        #pragma unroll
        for (int off = 16; off > 0; off >>= 1) s2 += __shfl_xor(s2, off, 32);
        __syncthreads();                                // tot reads done
        if ((tid & 31) == 0) red[tid >> 5] = s2;
        __syncthreads();
        float var = (red[0] + red[1] + red[2] + red[3]
                   + red[4] + red[5] + red[6] + red[7]) * (1.0f / HID);

        colsum += d * rsqrtf(var + LN_EPS) * g + bb;
        __syncthreads();                                // red reused next iter
    }

    float* mrow = molbuf + (size_t)m * KMOLP;
    mrow[tid] = colsum / (float)size;                   // cols 0..255
    if (tid < FEAT_DIM)
        mrow[HID + tid] = features[(size_t)m * FEAT_DIM + tid]; // 256..455
    if (tid < KMOLP - D_MOL_IN)
        mrow[D_MOL_IN + tid] = 0.0f;                    // 456..463
}

// ---------------- tiny output GEMM: out[2500,12] = molh @ Wm2 + bm2 ----------------
__global__ void ffn_out_kernel(const float* __restrict__ molh,
                               const float* __restrict__ Wm2,
                               const float* __restrict__ bm2,
                               float* __restrict__ out) {
    int gid = blockIdx.x * blockDim.x + threadIdx.x;
    if (gid >= N_MOLS * N_TASKS) return;
    int m = gid / N_TASKS;
    int t = gid % N_TASKS;
    float s = bm2[t];
    const float* hrow = molh + (size_t)m * FFN_HID;
    for (int k = 0; k < FFN_HID; ++k)
        s += hrow[k] * Wm2[(size_t)k * N_TASKS + t];
    out[gid] = s;
}

// ---------------- launcher ----------------
extern "C" void kernel_launch(void* const* d_in, const int* in_sizes, int n_in,
                              void* d_out, int out_size, void* d_ws, size_t ws_size,
                              hipStream_t stream) {
    const float* atom_output = (const float*)d_in[0];
    const float* f_atoms     = (const float*)d_in[1];
    const int*   a2a         = (const int*)  d_in[2];
    const int*   a_scope     = (const int*)  d_in[3];
    const float* features    = (const float*)d_in[4];
    const float* W1          = (const float*)d_in[5];
    const float* b1          = (const float*)d_in[6];
    const float* W2          = (const float*)d_in[7];
    const float* b2          = (const float*)d_in[8];
    const float* ln_g        = (const float*)d_in[9];
    const float* ln_b        = (const float*)d_in[10];
    const float* Wm1         = (const float*)d_in[11];
    const float* bm1         = (const float*)d_in[12];
    const float* Wm2         = (const float*)d_in[13];
    const float* bm2         = (const float*)d_in[14];
    float* out = (float*)d_out;

    // workspace layout (floats)
    float* ws = (float*)d_ws;
    size_t off = 0;
    float* xbuf  = ws + off; off += (size_t)N_ATOMS * K1P;       // 41.6 M
    float* W1p   = ws + off; off += (size_t)K1P * D_FF;          // 0.43 M
    float* h1    = ws + off; off += (size_t)N_ATOMS * D_FF;      // 102.4 M
    float* hpre  = ws + off; off += (size_t)N_ATOMS * HID;       // 25.6 M
    float* molb  = ws + off; off += (size_t)N_MOLS * KMOLP;      // 1.16 M
    float* Wm1p  = ws + off; off += (size_t)KMOLP * FFN_HID;     // 0.48 M
    float* molh  = ws + off; off += (size_t)N_MOLS * FFN_HID;    // 2.56 M
    (void)ws_size; (void)n_in; (void)in_sizes; (void)out_size;

    // 1) zero-pad weight K dims
    {
        int tot = K1P * D_FF;
        pad_rows_kernel<<<(tot + 255) / 256, 256, 0, stream>>>(W1, W1p, D_IN, K1P, D_FF);
    }
    {
        int tot = KMOLP * FFN_HID;
        pad_rows_kernel<<<(tot + 255) / 256, 256, 0, stream>>>(Wm1, Wm1p, D_MOL_IN, KMOLP, FFN_HID);
    }

    // 2) gather neighbors + concat features
    gather_concat_kernel<<<N_ATOMS, HID, 0, stream>>>(atom_output, f_atoms, a2a, xbuf);

    // 3) h1 = relu(x @ W1 + b1)   [100000 x 1024]
    {
        dim3 grid((N_ATOMS + 31) / 32, D_FF / 256);
        gemm_wmma_kernel<true><<<grid, 256, 0, stream>>>(xbuf, W1p, b1, h1,
                                                         N_ATOMS, D_FF, K1P);
    }
    // 4) hpre = h1 @ W2 + b2      [100000 x 256]  (K=1024 already aligned)
    {
        dim3 grid((N_ATOMS + 31) / 32, HID / 256);
        gemm_wmma_kernel<false><<<grid, 256, 0, stream>>>(h1, W2, b2, hpre,
                                                          N_ATOMS, HID, D_FF);
    }
    // 5) LayerNorm + segment mean + concat features -> molb [2500 x 464]
    ln_segmean_kernel<<<N_MOLS, HID, 0, stream>>>(hpre, a_scope, ln_g, ln_b,
                                                  features, molb);
    // 6) molh = relu(mol @ Wm1 + bm1)  [2500 x 1024]
    {
        dim3 grid((N_MOLS + 31) / 32, FFN_HID / 256);
        gemm_wmma_kernel<true><<<grid, 256, 0, stream>>>(molb, Wm1p, bm1, molh,
                                                         N_MOLS, FFN_HID, KMOLP);
    }
    // 7) out = molh @ Wm2 + bm2   [2500 x 12]
    {
        int tot = N_MOLS * N_TASKS;
        ffn_out_kernel<<<(tot + 255) / 256, 256, 0, stream>>>(molh, Wm2, bm2, out);
    }
}